// MaskedNormalizedCrossCorrelation2d_11527692223165
// MI455X (gfx1250) — compile-verified
//
#include <hip/hip_runtime.h>
#include <hip/hip_bf16.h>

// Problem constants (from reference): b=8, c=1, H=W=128, p=9 -> Hp=Wp=120
#define BATCH 8
#define IMG   128
#define P     9
#define HP    120            // patches per row/col
#define NPATCH (HP * HP)     // 14400 patches per batch
#define PATCH_ELEMS (P * P)  // 81
#define EPSV 1e-5f

typedef __attribute__((ext_vector_type(2))) float v2f;
typedef __attribute__((ext_vector_type(8))) float v8f;
typedef __attribute__((ext_vector_type(4))) int   v4i;

// ---- gfx1250 async global->LDS path (guarded; falls back to plain loads) ----
#if defined(__has_builtin)
#if __has_builtin(__builtin_amdgcn_global_load_async_to_lds_b128)
#define HAVE_ASYNC_LDS 1
#endif
#if __has_builtin(__builtin_amdgcn_s_wait_asynccnt)
#define HAVE_WAIT_ASYNC 1
#endif
#endif

__device__ __forceinline__ void async_wait0() {
#if defined(HAVE_WAIT_ASYNC)
    __builtin_amdgcn_s_wait_asynccnt(0);
#else
    asm volatile("s_wait_asynccnt 0x0" ::: "memory");
#endif
}

#if defined(HAVE_ASYNC_LDS)
// Builtin signature (from clang diagnostic): v4i pointers, global + LDS AS.
typedef __attribute__((address_space(1))) v4i* g4_t;
typedef __attribute__((address_space(3))) v4i* l4_t;
// One instruction copies 16B global -> LDS per lane, no VGPR round-trip,
// tracked with ASYNCcnt (gfx1250 GLOBAL_LOAD_ASYNC_TO_LDS_B128).
__device__ __forceinline__ void async_copy_b128(const float* g, float* l) {
    float* gn = const_cast<float*>(g);
    __builtin_amdgcn_global_load_async_to_lds_b128((g4_t)gn, (l4_t)l,
                                                   /*offset=*/0, /*cpol=*/0);
}
#endif

// Wave32 reduction on the matrix pipe.
// V_WMMA_F32_16X16X4_F32 with B = all-ones gives D[m][n] = sum_k A[m][k].
// Lane partial in A component 0, zero in component 1 -> rowsum_m = s_m + s_{m+16};
// D columns identical, so summing the 8 D vgprs gives a uniform half-total per
// half-wave; a second WMMA folds the halves -> full wave sum in every lane.
__device__ __forceinline__ float wave_reduce_wmma(float s) {
    v2f a;    a.x = s;    a.y = 0.0f;
    v2f ones; ones.x = 1.0f; ones.y = 1.0f;
    v8f c = {};
    c = __builtin_amdgcn_wmma_f32_16x16x4_f32(false, a, false, ones,
                                              (short)0, c, false, false);
    float t = c[0] + c[1] + c[2] + c[3] + c[4] + c[5] + c[6] + c[7];
    v2f a2; a2.x = t; a2.y = 0.0f;
    v8f c2 = {};
    c2 = __builtin_amdgcn_wmma_f32_16x16x4_f32(false, a2, false, ones,
                                               (short)0, c2, false, false);
    return c2[0];
}

// One block per (patch-row i, batch b). 128 threads; threads 0..119 each own
// one patch column j. Everything the block touches is staged in LDS; the
// 37 MB mask stream is read exactly once, fully coalesced.
__global__ __launch_bounds__(128)
void ncc_rows_kernel(const float* __restrict__ x1,
                     const float* __restrict__ x2,
                     const float* __restrict__ mask,
                     float* __restrict__ partials) {
    const int i   = blockIdx.x;   // patch row, 0..119
    const int b   = blockIdx.y;   // batch, 0..7
    const int tid = threadIdx.x;  // 0..127

    __shared__ __align__(16) float s_x1[P * IMG];         // rows i..i+8 of x1
    __shared__ __align__(16) float s_x2[P * IMG];         // rows i..i+8 of x2
    __shared__ __align__(16) float s_m[HP * PATCH_ELEMS]; // 120 masks of 81
    __shared__ float s_part[4];

    const float* img1 = x1 + (size_t)b * (IMG * IMG) + (size_t)i * IMG;
    const float* img2 = x2 + (size_t)b * (IMG * IMG) + (size_t)i * IMG;
    const float* mrow = mask + ((size_t)b * NPATCH + (size_t)i * HP) * PATCH_ELEMS;

#if defined(HAVE_ASYNC_LDS)
    // 288 float4 per image, 2430 float4 for the mask block: all 16B-aligned,
    // streamed straight into LDS via the async data mover path.
    for (int v = tid; v < (P * IMG) / 4; v += 128) {
        async_copy_b128(img1 + v * 4, &s_x1[v * 4]);
        async_copy_b128(img2 + v * 4, &s_x2[v * 4]);
    }
    for (int v = tid; v < (HP * PATCH_ELEMS) / 4; v += 128) {
        async_copy_b128(mrow + v * 4, &s_m[v * 4]);
    }
    async_wait0();
#else
    for (int idx = tid; idx < P * IMG; idx += 128) {
        s_x1[idx] = img1[idx];
        s_x2[idx] = img2[idx];
    }
    for (int idx = tid; idx < HP * PATCH_ELEMS; idx += 128) {
        s_m[idx] = mrow[idx];
    }
#endif
    // Speculative prefetch of the next row's mask block into L2.
    if (i + 1 < HP) {
        __builtin_prefetch(mrow + HP * PATCH_ELEMS + tid * 64, 0, 1);
    }
    __syncthreads();

    float s;
    if (tid < HP) {
        float sa = 0.f, sa2 = 0.f, sb = 0.f, sb2 = 0.f, sab = 0.f;
        const float* mp = &s_m[tid * PATCH_ELEMS];  // stride 81: conflict-free
#pragma unroll
        for (int dy = 0; dy < P; ++dy) {
            const float* r1 = &s_x1[dy * IMG + tid];
            const float* r2 = &s_x2[dy * IMG + tid];
            const float* mm = mp + dy * P;
#pragma unroll
            for (int dx = 0; dx < P; ++dx) {
                float a  = r1[dx];
                float bb = r2[dx] * mm[dx];
                sa  += a;
                sa2  = fmaf(a, a, sa2);
                sb  += bb;
                sb2  = fmaf(bb, bb, sb2);
                sab  = fmaf(a, bb, sab);
            }
        }
        // dot(norm(a), norm(b)) = (Sab - N*mu_a*mu_b) / sqrt((va+eps)(vb+eps))
        const float inv81 = 1.0f / 81.0f;
        float mua = sa * inv81;
        float mub = sb * inv81;
        float va  = fmaf(-mua, mua, sa2 * inv81) + EPSV;
        float vb  = fmaf(-mub, mub, sb2 * inv81) + EPSV;
        s = fmaf(-sa, mub, sab) / sqrtf(va * vb);
    } else {
        s = 0.0f;  // idle lanes contribute zero; EXEC is all-1 again at the WMMA
    }

    float wsum = wave_reduce_wmma(s);       // all 128 threads execute this
    const int lane = tid & 31, wave = tid >> 5;
    if (lane == 0) s_part[wave] = wsum;
    __syncthreads();
    if (tid == 0) {
        partials[b * HP + i] = s_part[0] + s_part[1] + s_part[2] + s_part[3];
    }
}

// One block per batch: fold 120 row-partials -> final score.
__global__ __launch_bounds__(128)
void ncc_final_kernel(const float* __restrict__ partials,
                      float* __restrict__ out) {
    const int b   = blockIdx.x;
    const int tid = threadIdx.x;
    __shared__ float s_part[4];

    float p = (tid < HP) ? partials[b * HP + tid] : 0.0f;
    float wsum = wave_reduce_wmma(p);
    if ((tid & 31) == 0) s_part[tid >> 5] = wsum;
    __syncthreads();
    if (tid == 0) {
        const float scale = 1.0f / ((float)NPATCH * (float)PATCH_ELEMS); // 1/1166400
        out[b] = (s_part[0] + s_part[1] + s_part[2] + s_part[3]) * scale;
    }
}

extern "C" void kernel_launch(void* const* d_in, const int* in_sizes, int n_in,
                              void* d_out, int out_size, void* d_ws, size_t ws_size,
                              hipStream_t stream) {
    (void)in_sizes; (void)n_in; (void)out_size; (void)ws_size;
    const float* x1   = (const float*)d_in[0];  // [8,1,128,128]
    const float* x2   = (const float*)d_in[1];  // [8,1,128,128]
    const float* mask = (const float*)d_in[2];  // [8,14400,9,9]
    float* out      = (float*)d_out;            // [8]
    float* partials = (float*)d_ws;             // 960 floats of scratch

    dim3 grid(HP, BATCH);  // 120 x 8 = 960 blocks
    ncc_rows_kernel<<<grid, 128, 0, stream>>>(x1, x2, mask, partials);
    ncc_final_kernel<<<BATCH, 128, 0, stream>>>(partials, out);
}